// pLayer_54683523613165
// MI455X (gfx1250) — compile-verified
//
#include <hip/hip_runtime.h>
#include <hip/hip_bf16.h>

// ---------------------------------------------------------------------------
// Problem dims (from reference): F=2, V=4, E=2048, N_IN=512, M=514, N_OUT=512
// ---------------------------------------------------------------------------
constexpr int F_DIM = 2, V_DIM = 4, FV = F_DIM * V_DIM;
constexpr int E_DIM = 2048, NIN = 512, NOUT = 512, M_DIM = NIN + 2;  // 514
constexpr float GMIN = 0.01f, GMAX = 10.0f, EPS_TRAIN = 0.1f, PGMIN = 1e-4f;

// workspace layout (float offsets)
constexpr size_t OFF_WSIG = 0;                                   // Wsigned [FV, M, N]
constexpr size_t SZ_WSIG  = (size_t)FV * M_DIM * NOUT;           // 2,105,344
constexpr size_t OFF_GT   = OFF_WSIG + SZ_WSIG;                  // g_tilde [M, N]
constexpr size_t SZ_GT    = (size_t)M_DIM * NOUT;                // 263,168
constexpr size_t OFF_Z    = OFF_GT + SZ_GT;                      // Z [FV, E, N]
constexpr size_t SZ_Z     = (size_t)FV * E_DIM * NOUT;           // 8,388,608
constexpr size_t OFF_GRS  = OFF_Z + SZ_Z;                        // grs [M] (row sums of g_tilde)
constexpr size_t SZ_GRS   = 528;                                 // padded
constexpr size_t OFF_G    = OFF_GRS + SZ_GRS;                    // G [N] (col sums of g_tilde)
constexpr size_t SZ_G     = 512;
constexpr size_t OFF_ACTP = OFF_G + SZ_G;                        // act params float4 per n
constexpr size_t SZ_ACTP  = (size_t)NOUT * 4;
constexpr size_t OFF_CS   = OFF_ACTP + SZ_ACTP;                  // colsum(z) [FV, N]
constexpr size_t SZ_CS    = (size_t)FV * NOUT;                   // 4096
constexpr size_t OFF_ACC  = OFF_CS + SZ_CS;                      // acc[0]=p1 acc[1]=p2raw acc[2]=p3
constexpr size_t SZ_ACC   = 16;

__constant__ float c_etas[4][4] = {
    {0.05f, 0.90f, 0.20f, 8.0f},
    {0.10f, 0.80f, 0.30f, 5.0f},
    {0.00f, 1.00f, 0.25f, 10.0f},
    {0.15f, 0.70f, 0.15f, 6.0f}};

typedef __attribute__((ext_vector_type(2))) float v2f;
typedef __attribute__((ext_vector_type(8))) float v8f;

__device__ inline float wave_reduce_sum(float v) {
  #pragma unroll
  for (int o = 16; o; o >>= 1) v += __shfl_xor(v, o, 32);
  return v;
}

// ---------------------------------------------------------------------------
// Kernel 1: per output column n -> theta transform, g_tilde, G, activation
// selection (gumbel argmax), Wsigned for each of the 8 (f,v) noise draws.
// grid = NOUT blocks, block = 128 threads
// ---------------------------------------------------------------------------
__global__ __launch_bounds__(128) void prep_kernel(
    const float* __restrict__ theta_, const float* __restrict__ noise_u,
    const float* __restrict__ cof, const float* __restrict__ gum,
    float* __restrict__ Wsig, float* __restrict__ gt, float* __restrict__ G,
    float* __restrict__ actp) {
  const int n = blockIdx.x;
  const int tid = threadIdx.x;
  __shared__ float sthc[M_DIM];   // clamped theta column
  __shared__ float stn[M_DIM];    // theta_noisy column (per fv)
  __shared__ float red[128];

  for (int m = tid; m < M_DIM; m += 128) {
    float t = theta_[(size_t)m * NOUT + n];
    sthc[m] = fminf(fmaxf(t, -GMAX), GMAX);
  }
  __syncthreads();

  // column min and column sum of g_init = |clamp(theta)|
  float mn = 1e30f, sm = 0.0f;
  for (int m = tid; m < M_DIM; m += 128) {
    float g = fabsf(sthc[m]);
    mn = fminf(mn, g);
    sm += g;
  }
  red[tid] = mn; __syncthreads();
  for (int s = 64; s > 0; s >>= 1) { if (tid < s) red[tid] = fminf(red[tid], red[tid + s]); __syncthreads(); }
  const float colmin = red[0]; __syncthreads();
  red[tid] = sm; __syncthreads();
  for (int s = 64; s > 0; s >>= 1) { if (tid < s) red[tid] += red[tid + s]; __syncthreads(); }
  const float gsum = red[0]; __syncthreads();

  const float scale = PGMIN / colmin;
  for (int m = tid; m < M_DIM; m += 128)
    gt[(size_t)m * NOUT + n] = fabsf(sthc[m]) * scale;
  if (tid == 0) G[n] = gsum * scale;

  // per-(f,v): theta_noisy, normalization denominator, Wsigned
  for (int fv = 0; fv < FV; ++fv) {
    float part = 0.0f;
    for (int m = tid; m < M_DIM; m += 128) {
      float c = sthc[m];
      float tht = (fabsf(c) < GMIN) ? 0.0f : c;              // thresholded theta
      float nu = noise_u[((size_t)fv * M_DIM + m) * NOUT + n];
      float tn = tht * ((nu * 2.0f - 1.0f) * EPS_TRAIN + 1.0f);
      stn[m] = tn;
      part += fabsf(tn);
    }
    red[tid] = part; __syncthreads();
    for (int s = 64; s > 0; s >>= 1) { if (tid < s) red[tid] += red[tid + s]; __syncthreads(); }
    const float denom = red[0] + 1e-10f;
    for (int m = tid; m < M_DIM; m += 128)
      Wsig[((size_t)fv * M_DIM + m) * NOUT + n] = stn[m] / denom;
    __syncthreads();
  }

  // activation selection: argmax over i of clip(c)+gumbel  == argmax of softmax
  if (tid == 0) {
    float best = -1e30f; int bi = 0;
    #pragma unroll
    for (int i = 0; i < 4; ++i) {
      float ci = fminf(fmaxf(cof[(size_t)i * NOUT + n], -1.0f), 1.0f);
      float gu = gum[(size_t)i * NOUT + n];
      float gn = -logf(-logf(gu + 1e-20f) + 1e-20f);
      float s = ci + gn;
      if (s > best) { best = s; bi = i; }                    // first-max tie like argmax
    }
    actp[(size_t)n * 4 + 0] = c_etas[bi][0];
    actp[(size_t)n * 4 + 1] = c_etas[bi][1];
    actp[(size_t)n * 4 + 2] = c_etas[bi][2];
    actp[(size_t)n * 4 + 3] = c_etas[bi][3];
  }
}

// ---------------------------------------------------------------------------
// Kernel 2: grs[m] = sum_n g_tilde[m,n].  grid = M_DIM, block = 128
// ---------------------------------------------------------------------------
__global__ __launch_bounds__(128) void grs_kernel(const float* __restrict__ gt,
                                                  float* __restrict__ grs) {
  const int m = blockIdx.x;
  __shared__ float red[128];
  float s = 0.0f;
  for (int n = threadIdx.x; n < NOUT; n += 128) s += gt[(size_t)m * NOUT + n];
  red[threadIdx.x] = s; __syncthreads();
  for (int o = 64; o > 0; o >>= 1) { if (threadIdx.x < o) red[threadIdx.x] += red[threadIdx.x + o]; __syncthreads(); }
  if (threadIdx.x == 0) grs[m] = red[0];
}

// ---------------------------------------------------------------------------
// Kernel 3: p1 partial = sum over all a_prev elements of a^2 * grs[m]
// (sx2==sn2, pos+neg==1 collapses both einsums). grid-stride.
// ---------------------------------------------------------------------------
__global__ __launch_bounds__(256) void p1_kernel(const float* __restrict__ X,
                                                 const float* __restrict__ grs,
                                                 float* __restrict__ acc) {
  const size_t total = (size_t)FV * E_DIM * NIN;
  __shared__ float red[256];
  float loc = 0.0f;
  for (size_t i = (size_t)blockIdx.x * blockDim.x + threadIdx.x; i < total;
       i += (size_t)gridDim.x * blockDim.x) {
    int m = (int)(i & (NIN - 1));
    float v = X[i];
    loc += v * v * grs[m];
  }
  red[threadIdx.x] = loc; __syncthreads();
  for (int o = 128; o > 0; o >>= 1) { if (threadIdx.x < o) red[threadIdx.x] += red[threadIdx.x + o]; __syncthreads(); }
  if (threadIdx.x == 0) atomicAdd(&acc[0], red[0]);
}

// ---------------------------------------------------------------------------
// Kernel 4: z = X @ Wsigned (+bias row 512), fused activation, p3 partial,
// column sums of z (== C[.,512,.]).  f32 WMMA 16x16x4.
// grid = (N/64, E/64, FV), block = 128 (4 waves); each wave: 16e x 64n.
// ---------------------------------------------------------------------------
__global__ __launch_bounds__(128) void zgemm_kernel(
    const float* __restrict__ X, const float* __restrict__ Wsig,
    const float* __restrict__ G, const float* __restrict__ actp,
    float* __restrict__ Z, float* __restrict__ Anew,
    float* __restrict__ colsum, float* __restrict__ acc) {
  const int fv = blockIdx.z;
  const int e0 = blockIdx.y * 64;
  const int n0 = blockIdx.x * 64;
  const int wave = threadIdx.x >> 5, lane = threadIdx.x & 31;
  const int eW = e0 + wave * 16;
  const int lm = lane & 15;            // M (=e row) for A, N for B/C
  const int kb = (lane >> 4) << 1;     // K base: 0 for lanes 0-15, 2 for 16-31

  const float* Xf = X + (size_t)fv * E_DIM * NIN;
  const float* Wf = Wsig + (size_t)fv * M_DIM * NOUT;

  v8f accv[4] = {v8f{}, v8f{}, v8f{}, v8f{}};
  for (int m0 = 0; m0 < NIN; m0 += 4) {               // K loop over m (512, mult of 4)
    v2f a;                                            // A[e,m] 16x4 tile
    a.x = Xf[(size_t)(eW + lm) * NIN + m0 + kb];
    a.y = Xf[(size_t)(eW + lm) * NIN + m0 + kb + 1];
    #pragma unroll
    for (int nt = 0; nt < 4; ++nt) {
      const int n = n0 + nt * 16 + lm;
      v2f b;                                          // B[m,n] 4x16 tile
      b.x = Wf[(size_t)(m0 + kb) * NOUT + n];
      b.y = Wf[(size_t)(m0 + kb + 1) * NOUT + n];
      accv[nt] = __builtin_amdgcn_wmma_f32_16x16x4_f32(
          false, a, false, b, (short)0, accv[nt], false, false);
    }
  }

  float p3loc = 0.0f;
  #pragma unroll
  for (int nt = 0; nt < 4; ++nt) {
    const int n = n0 + nt * 16 + lm;
    const float bias = Wf[(size_t)512 * NOUT + n];    // x_ext ones column
    const float4 pr = ((const float4*)actp)[n];
    const float gn = G[n];
    float cs = 0.0f;
    #pragma unroll
    for (int j = 0; j < 8; ++j) {                     // C/D layout: M=j(+8), N=lm
      const int e = eW + j + ((lane < 16) ? 0 : 8);
      const float z = accv[nt][j] + bias;
      const size_t idx = ((size_t)fv * E_DIM + e) * NOUT + n;
      Z[idx] = z;
      Anew[idx] = pr.x + pr.y * tanhf((z - pr.z) * pr.w);
      p3loc += z * z * gn;
      cs += z;
    }
    cs += __shfl_xor(cs, 16, 32);                     // lanes L and L+16 share n
    if (lane < 16) atomicAdd(&colsum[(size_t)fv * NOUT + n], cs);
  }
  p3loc = wave_reduce_sum(p3loc);
  if (lane == 0) atomicAdd(&acc[2], p3loc);
}

// ---------------------------------------------------------------------------
// Kernel 5: C = X^T @ Z per (f,v), fused reduction sum C*sign(Wsig)*g_tilde
// into p2raw (p2 = -2*p2raw).  Only m in [0,512); m=512 handled in finalize.
// grid = (N/64, NIN/16, FV), block = 128; each wave: 16m x 16n, K=E=2048.
// ---------------------------------------------------------------------------
__global__ __launch_bounds__(128) void cgemm_kernel(
    const float* __restrict__ X, const float* __restrict__ Wsig,
    const float* __restrict__ gt, const float* __restrict__ Z,
    float* __restrict__ acc) {
  const int fv = blockIdx.z;
  const int m0 = blockIdx.y * 16;
  const int wave = threadIdx.x >> 5, lane = threadIdx.x & 31;
  const int n0 = blockIdx.x * 64 + wave * 16;
  const int lm = lane & 15;
  const int kb = (lane >> 4) << 1;

  const float* Xf = X + (size_t)fv * E_DIM * NIN;
  const float* Zf = Z + (size_t)fv * E_DIM * NOUT;

  v8f accv = {};
  #pragma unroll 4
  for (int e = 0; e < E_DIM; e += 4) {                // K loop over e
    v2f a;                                            // A = X^T tile: A[m, e]
    a.x = Xf[(size_t)(e + kb) * NIN + m0 + lm];
    a.y = Xf[(size_t)(e + kb + 1) * NIN + m0 + lm];
    v2f b;                                            // B = Z tile: B[e, n]
    b.x = Zf[(size_t)(e + kb) * NOUT + n0 + lm];
    b.y = Zf[(size_t)(e + kb + 1) * NOUT + n0 + lm];
    accv = __builtin_amdgcn_wmma_f32_16x16x4_f32(
        false, a, false, b, (short)0, accv, false, false);
  }

  const float* Wf = Wsig + (size_t)fv * M_DIM * NOUT;
  float loc = 0.0f;
  #pragma unroll
  for (int j = 0; j < 8; ++j) {
    const int m = m0 + j + ((lane < 16) ? 0 : 8);
    const int n = n0 + lm;
    const float w = Wf[(size_t)m * NOUT + n];
    const float s = (w >= 0.0f) ? 1.0f : -1.0f;       // sign(theta_noisy)
    loc += accv[j] * s * gt[(size_t)m * NOUT + n];
  }
  loc = wave_reduce_sum(loc);
  if (lane == 0) atomicAdd(&acc[1], loc);
}

// ---------------------------------------------------------------------------
// Kernel 6: finalize mac_power.
// ---------------------------------------------------------------------------
__global__ __launch_bounds__(128) void finalize_kernel(
    const float* __restrict__ Wsig, const float* __restrict__ gt,
    const float* __restrict__ grs, const float* __restrict__ colsum,
    const float* __restrict__ acc, float* __restrict__ out_power) {
  __shared__ float red[128];
  float loc = 0.0f;
  for (int i = threadIdx.x; i < FV * NOUT; i += 128) { // C[.,512,.] = colsum(z)
    const int fv = i >> 9, n = i & (NOUT - 1);
    const float w = Wsig[((size_t)fv * M_DIM + 512) * NOUT + n];
    const float s = (w >= 0.0f) ? 1.0f : -1.0f;
    loc += colsum[i] * s * gt[(size_t)512 * NOUT + n];
  }
  red[threadIdx.x] = loc; __syncthreads();
  for (int o = 64; o > 0; o >>= 1) { if (threadIdx.x < o) red[threadIdx.x] += red[threadIdx.x + o]; __syncthreads(); }
  if (threadIdx.x == 0) {
    const float p1 = acc[0] + (float)FV * (float)E_DIM * grs[512]; // ones column of sx2
    const float p2 = -2.0f * (acc[1] + red[0]);
    const float p3 = acc[2];
    *out_power = (p1 + p2 + p3) / (float)(E_DIM * V_DIM * F_DIM);
  }
}

// ---------------------------------------------------------------------------
extern "C" void kernel_launch(void* const* d_in, const int* in_sizes, int n_in,
                              void* d_out, int out_size, void* d_ws, size_t ws_size,
                              hipStream_t stream) {
  const float* a_prev = (const float*)d_in[0];   // [F,V,E,N_IN]
  const float* theta_ = (const float*)d_in[1];   // [M,N]
  const float* cof    = (const float*)d_in[2];   // [4,N]
  const float* nois   = (const float*)d_in[3];   // [F,V,M,N]
  const float* gum    = (const float*)d_in[4];   // [4,N]
  float* out = (float*)d_out;                    // a_new [F,V,E,N] ++ mac_power

  float* ws   = (float*)d_ws;
  float* Wsig = ws + OFF_WSIG;
  float* gt   = ws + OFF_GT;
  float* Z    = ws + OFF_Z;
  float* grs  = ws + OFF_GRS;
  float* G    = ws + OFF_G;
  float* actp = ws + OFF_ACTP;
  float* cs   = ws + OFF_CS;
  float* acc  = ws + OFF_ACC;

  // zero the atomically-accumulated regions (colsum + acc, contiguous)
  hipMemsetAsync(cs, 0, (SZ_CS + SZ_ACC) * sizeof(float), stream);

  prep_kernel<<<NOUT, 128, 0, stream>>>(theta_, nois, cof, gum, Wsig, gt, G, actp);
  grs_kernel<<<M_DIM, 128, 0, stream>>>(gt, grs);
  p1_kernel<<<2048, 256, 0, stream>>>(a_prev, grs, acc);
  zgemm_kernel<<<dim3(NOUT / 64, E_DIM / 64, FV), 128, 0, stream>>>(
      a_prev, Wsig, G, actp, Z, out, cs, acc);
  cgemm_kernel<<<dim3(NOUT / 64, NIN / 16, FV), 128, 0, stream>>>(
      a_prev, Wsig, gt, Z, acc);
  finalize_kernel<<<1, 128, 0, stream>>>(Wsig, gt, grs, cs, acc,
                                         out + (size_t)FV * E_DIM * NOUT);
}